// TrendHead_12137577578983
// MI455X (gfx1250) — compile-verified
//
#include <hip/hip_runtime.h>

// ---------------------------------------------------------------------------
// TrendHead: degree-2 polynomial detrend, x[32,4096,128] f32.
// out = [low | x-low | zeros | zeros], each plane B*T*C floats.
// Memory-bound: 64MB read + 256MB write -> ~13.7us floor @ 23.3 TB/s.
// Pass1: per-(b,c) raw moments M_k = sum_t t^k x  (atomic f32 accumulation).
// Host:  3x3 Gram inverse in double (pinv(T) = Ginv @ T^T).
// Pass2: low tile = Tmat[16x4] @ coef[4x16] via V_WMMA_F32_16X16X4_F32,
//        NT stores for the 256MB write-only output.
// ---------------------------------------------------------------------------

typedef float v2f __attribute__((ext_vector_type(2)));
typedef float v4f __attribute__((ext_vector_type(4)));
typedef float v8f __attribute__((ext_vector_type(8)));

namespace {
constexpr int B_ = 32;
constexpr int T_ = 4096;
constexpr int C_ = 128;
constexpr long long NPLANE = (long long)B_ * T_ * C_;   // 16,777,216
constexpr int MOM_TCHUNKS = 64;                          // pass-1 T split
constexpr int MOM_TLEN    = T_ / MOM_TCHUNKS;            // 64
}

struct GinvArgs { float g[9]; };   // row-major 3x3, coef = Ginv * moments

// ---- zero the 48KB moment scratch (ws is poisoned, must self-init) --------
__global__ void zero_moments_kernel(float* __restrict__ mom) {
    int i = blockIdx.x * blockDim.x + threadIdx.x;
    if (i < B_ * 3 * C_) mom[i] = 0.0f;
}

// ---- pass 1: raw moments, coalesced over C, chunked over T ----------------
__global__ __launch_bounds__(C_) void moments_kernel(const float* __restrict__ x,
                                                     float* __restrict__ mom) {
    const int b     = blockIdx.y;     // 0..31
    const int chunk = blockIdx.x;     // 0..63
    const int c     = threadIdx.x;    // 0..127
    const float invTm1 = 1.0f / (float)(T_ - 1);

    const int t0 = chunk * MOM_TLEN;
    const float* p = x + ((long long)b * T_ + t0) * C_ + c;

    float m0 = 0.f, m1 = 0.f, m2 = 0.f;
    #pragma unroll 4
    for (int i = 0; i < MOM_TLEN; ++i) {
        float v  = p[(long long)i * C_];
        float tn = (float)(t0 + i) * invTm1;
        m0 += v;
        m1 = fmaf(tn, v, m1);
        m2 = fmaf(tn * tn, v, m2);
    }
    float* mb = mom + b * 3 * C_;
    atomicAdd(mb + 0 * C_ + c, m0);
    atomicAdd(mb + 1 * C_ + c, m1);
    atomicAdd(mb + 2 * C_ + c, m2);
}

// ---- pass 2: WMMA rank-3 reconstruction + residual ------------------------
// One wave per (b, 16-channel tile, 128-t chunk); 8 WMMA tiles per wave.
// A (16x4 f32, ISA layout): lanes 0-15 -> {K0=1, K1=t}; lanes 16-31 -> {K2=t^2, K3=0}
// B (4x16 f32, layout mirrors C/D striping: VGPR v holds rows v and v+2):
//   lanes 0-15 -> {c0, c1}; lanes 16-31 -> {c2, 0}
// D (16x16 f32): VGPR v -> row (v + 8*laneHalf), col = lane&15.
__global__ __launch_bounds__(128) void recon_kernel(const float* __restrict__ x,
                                                    const float* __restrict__ mom,
                                                    float* __restrict__ out,
                                                    GinvArgs ga) {
    const int lane       = threadIdx.x & 31;
    const int waveGlobal = blockIdx.x * 4 + (threadIdx.x >> 5);   // 0..8191
    const int b      = waveGlobal >> 8;          // 32
    const int rem    = waveGlobal & 255;
    const int tchunk = rem >> 3;                 // 32 chunks of 128 t
    const int ctile  = rem & 7;                  // 8 tiles of 16 c
    const int n = lane & 15;
    const int h = lane >> 4;
    const int c = ctile * 16 + n;
    const float invTm1 = 1.0f / (float)(T_ - 1);

    // per-channel polynomial coefficients: coef = Ginv * M
    const float* mb = mom + b * 3 * C_;
    const float M0 = mb[0 * C_ + c];
    const float M1 = mb[1 * C_ + c];
    const float M2 = mb[2 * C_ + c];
    const float c0 = ga.g[0] * M0 + ga.g[1] * M1 + ga.g[2] * M2;
    const float c1 = ga.g[3] * M0 + ga.g[4] * M1 + ga.g[5] * M2;
    const float c2 = ga.g[6] * M0 + ga.g[7] * M1 + ga.g[8] * M2;

    v2f bmat;
    bmat.x = h ? c2  : c0;   // VGPR0: K=0 (low lanes) / K=2 (high lanes)
    bmat.y = h ? 0.f : c1;   // VGPR1: K=1 (low lanes) / K=3 (high lanes)

    float* __restrict__ lowOut = out;
    float* __restrict__ resOut = out + NPLANE;

    #pragma unroll
    for (int it = 0; it < 8; ++it) {
        const int tb = tchunk * 128 + it * 16;
        const float tn = (float)(tb + n) * invTm1;
        v2f amat;
        amat.x = h ? tn * tn : 1.0f;   // K=0 / K=2
        amat.y = h ? 0.f     : tn;     // K=1 / K=3

        v8f acc = {};
        acc = __builtin_amdgcn_wmma_f32_16x16x4_f32(
                  false, amat, false, bmat, (short)0, acc, false, false);

        #pragma unroll
        for (int v = 0; v < 8; ++v) {
            const int t = tb + v + h * 8;
            const long long idx = ((long long)b * T_ + t) * C_ + c;
            const float low = acc[v];
            const float xv  = x[idx];                       // L2 hit (64MB < 192MB L2)
            __builtin_nontemporal_store(low,      &lowOut[idx]);
            __builtin_nontemporal_store(xv - low, &resOut[idx]);
        }
    }
}

// ---- zero planes 2 and 3 with wide NT stores ------------------------------
__global__ void zeros_kernel(v4f* __restrict__ out4, long long n4) {
    long long i = (long long)blockIdx.x * blockDim.x + threadIdx.x;
    if (i < n4) {
        v4f z = {0.f, 0.f, 0.f, 0.f};
        __builtin_nontemporal_store(z, &out4[i]);
    }
}

extern "C" void kernel_launch(void* const* d_in, const int* in_sizes, int n_in,
                              void* d_out, int out_size, void* d_ws, size_t ws_size,
                              hipStream_t stream) {
    (void)in_sizes; (void)n_in; (void)out_size; (void)ws_size;
    const float* x   = (const float*)d_in[0];
    float*       out = (float*)d_out;
    float*       mom = (float*)d_ws;            // 32*3*128 floats = 48KB

    // Host-side: Gram matrix G = T^T T for basis [1, t, t^2], t = j/(T-1),
    // inverted in double for a well-conditioned closed-form pinv.
    double S[5] = {0, 0, 0, 0, 0};
    for (int j = 0; j < T_; ++j) {
        double t = (double)j / (double)(T_ - 1);
        double p = 1.0;
        for (int k = 0; k < 5; ++k) { S[k] += p; p *= t; }
    }
    double G[3][3] = {{S[0], S[1], S[2]}, {S[1], S[2], S[3]}, {S[2], S[3], S[4]}};
    double det = G[0][0] * (G[1][1] * G[2][2] - G[1][2] * G[2][1])
               - G[0][1] * (G[1][0] * G[2][2] - G[1][2] * G[2][0])
               + G[0][2] * (G[1][0] * G[2][1] - G[1][1] * G[2][0]);
    double inv[3][3];
    inv[0][0] =  (G[1][1] * G[2][2] - G[1][2] * G[2][1]) / det;
    inv[0][1] = -(G[0][1] * G[2][2] - G[0][2] * G[2][1]) / det;
    inv[0][2] =  (G[0][1] * G[1][2] - G[0][2] * G[1][1]) / det;
    inv[1][0] = -(G[1][0] * G[2][2] - G[1][2] * G[2][0]) / det;
    inv[1][1] =  (G[0][0] * G[2][2] - G[0][2] * G[2][0]) / det;
    inv[1][2] = -(G[0][0] * G[1][2] - G[0][2] * G[1][0]) / det;
    inv[2][0] =  (G[1][0] * G[2][1] - G[1][1] * G[2][0]) / det;
    inv[2][1] = -(G[0][0] * G[2][1] - G[0][1] * G[2][0]) / det;
    inv[2][2] =  (G[0][0] * G[1][1] - G[0][1] * G[1][0]) / det;

    GinvArgs ga;
    for (int r = 0; r < 3; ++r)
        for (int cc = 0; cc < 3; ++cc)
            ga.g[r * 3 + cc] = (float)inv[r][cc];

    // 0) zero moment scratch (48KB)
    zero_moments_kernel<<<(B_ * 3 * C_ + 255) / 256, 256, 0, stream>>>(mom);

    // 1) raw moments: grid (64 T-chunks, 32 batches) x 128 channels
    moments_kernel<<<dim3(MOM_TCHUNKS, B_), C_, 0, stream>>>(x, mom);

    // 2) WMMA reconstruction + residual: 8192 waves, 4 wave32 per block
    recon_kernel<<<2048, 128, 0, stream>>>(x, mom, out, ga);

    // 3) zero planes 2 and 3 (2 * NPLANE floats) with b128 NT stores
    const long long n4 = (2 * NPLANE) / 4;   // 8,388,608 float4
    zeros_kernel<<<(unsigned)((n4 + 255) / 256), 256, 0, stream>>>(
        (v4f*)(out + 2 * NPLANE), n4);
}